// EproPnPLossWrapper_10187662426468
// MI455X (gfx1250) — compile-verified
//
#include <hip/hip_runtime.h>
#include <math.h>

#define NPTS 400
#define LM_ITERS_N 5
#define RS_PTS 16
#define RS_PROPS 4
#define RS_ITERS_N 3
#define MC_ITERS_N 4
#define MC_PER 100
#define NSAMP (MC_ITERS_N * MC_PER)
#define REL_DELTA 0.1f
#define LOG2PI_F 1.8378770664093453f
#define TPB 256
#define NWAVES 8
#define PPW (NPTS / NWAVES)   /* 50 points per wave  */
#define ROWS_PW (2 * PPW)     /* 100 J-rows per wave */

typedef __attribute__((ext_vector_type(2))) float v2f;
typedef __attribute__((ext_vector_type(8))) float v8f;

// ---------------------------------------------------------------- Jet (6-dim forward AD)
struct Jet { float v; float g[6]; };

__device__ inline Jet jc(float x) {
  Jet r; r.v = x;
#pragma unroll
  for (int i = 0; i < 6; ++i) r.g[i] = 0.f;
  return r;
}
__device__ inline Jet operator+(const Jet& a, const Jet& b) {
  Jet r; r.v = a.v + b.v;
#pragma unroll
  for (int i = 0; i < 6; ++i) r.g[i] = a.g[i] + b.g[i];
  return r;
}
__device__ inline Jet operator-(const Jet& a, const Jet& b) {
  Jet r; r.v = a.v - b.v;
#pragma unroll
  for (int i = 0; i < 6; ++i) r.g[i] = a.g[i] - b.g[i];
  return r;
}
__device__ inline Jet operator*(const Jet& a, const Jet& b) {
  Jet r; r.v = a.v * b.v;
#pragma unroll
  for (int i = 0; i < 6; ++i) r.g[i] = a.v * b.g[i] + b.v * a.g[i];
  return r;
}
__device__ inline Jet operator*(const Jet& a, float s) {
  Jet r; r.v = a.v * s;
#pragma unroll
  for (int i = 0; i < 6; ++i) r.g[i] = a.g[i] * s;
  return r;
}
__device__ inline Jet operator+(const Jet& a, float s) { Jet r = a; r.v += s; return r; }
__device__ inline Jet operator-(const Jet& a, float s) { Jet r = a; r.v -= s; return r; }
__device__ inline Jet jdiv(const Jet& a, const Jet& b) {
  float inv = 1.f / b.v;
  Jet r; r.v = a.v * inv;
#pragma unroll
  for (int i = 0; i < 6; ++i) r.g[i] = (a.g[i] - r.v * b.g[i]) * inv;
  return r;
}
__device__ inline Jet jsqrt(const Jet& a) {
  float s = sqrtf(a.v);
  Jet r; r.v = s;
  float k = 0.5f / fmaxf(s, 1e-20f);
#pragma unroll
  for (int i = 0; i < 6; ++i) r.g[i] = k * a.g[i];
  return r;
}
__device__ inline Jet jmaxc(const Jet& a, float c) { return (a.v >= c) ? a : jc(c); }
__device__ inline Jet jminc(const Jet& a, float c) { return (a.v <= c) ? a : jc(c); }

// ---------------------------------------------------------------- shared memory
struct SM {
  float x3[NPTS][3];
  float x2[NPTS][2];
  float w2[NPTS][2];
  float Jt[NWAVES][8][ROWS_PW];    // per-wave J, COLUMN-major: cols 0..5 = J, col 6 = r, col 7 = 0
  float Hpart[NWAVES][7][7];
  float Hsum[7][7];
  float red[TPB];
  float dall[NSAMP][6];
  float scost[NSAMP];
  float lw[NSAMP];
  float mus[MC_ITERS_N][6];
  float sigs[MC_ITERS_N][6];
  float cand_t[RS_PROPS][3];
  float cand_q[RS_PROPS][4];
};

// ---------------------------------------------------------------- block reductions
__device__ float block_sum(SM* sm, float v) {
  int t = threadIdx.x;
  sm->red[t] = v; __syncthreads();
  for (int s = TPB / 2; s > 0; s >>= 1) {
    if (t < s) sm->red[t] += sm->red[t + s];
    __syncthreads();
  }
  float r = sm->red[0]; __syncthreads();
  return r;
}
__device__ float block_max(SM* sm, float v) {
  int t = threadIdx.x;
  sm->red[t] = v; __syncthreads();
  for (int s = TPB / 2; s > 0; s >>= 1) {
    if (t < s) sm->red[t] = fmaxf(sm->red[t], sm->red[t + s]);
    __syncthreads();
  }
  float r = sm->red[0]; __syncthreads();
  return r;
}

// ---------------------------------------------------------------- geometry (primal)
__device__ void quat_to_rot(const float* q, float* R) {
  float w = q[0], x = q[1], y = q[2], z = q[3];
  R[0] = 1.f - 2.f * (y * y + z * z); R[1] = 2.f * (x * y - w * z); R[2] = 2.f * (x * z + w * y);
  R[3] = 2.f * (x * y + w * z); R[4] = 1.f - 2.f * (x * x + z * z); R[5] = 2.f * (y * z - w * x);
  R[6] = 2.f * (x * z - w * y); R[7] = 2.f * (y * z + w * x); R[8] = 1.f - 2.f * (x * x + y * y);
}
__device__ void quat_mul4(const float* a, const float* b, float* o) {
  o[0] = a[0] * b[0] - a[1] * b[1] - a[2] * b[2] - a[3] * b[3];
  o[1] = a[0] * b[1] + a[1] * b[0] + a[2] * b[3] - a[3] * b[2];
  o[2] = a[0] * b[2] - a[1] * b[3] + a[2] * b[0] + a[3] * b[1];
  o[3] = a[0] * b[3] + a[1] * b[2] - a[2] * b[1] + a[3] * b[0];
}
__device__ void normalize4(float* q) {
  float n = sqrtf(q[0] * q[0] + q[1] * q[1] + q[2] * q[2] + q[3] * q[3]);
  float inv = 1.f / fmaxf(n, 1e-20f);
#pragma unroll
  for (int i = 0; i < 4; ++i) q[i] *= inv;
}
__device__ void rotvec_to_quat(const float* v, float* q) {
  float ang = sqrtf(v[0] * v[0] + v[1] * v[1] + v[2] * v[2]);
  float half_a = 0.5f * ang;
  float s = (ang > 1e-8f) ? (sinf(half_a) / fmaxf(ang, 1e-8f)) : 0.5f;
  q[0] = cosf(half_a); q[1] = v[0] * s; q[2] = v[1] * s; q[3] = v[2] * s;
}
__device__ void rot_to_quat(const float* R, float* q) {
  float m00 = R[0], m01 = R[1], m02 = R[2];
  float m10 = R[3], m11 = R[4], m12 = R[5];
  float m20 = R[6], m21 = R[7], m22 = R[8];
  float tr = m00 + m11 + m22;
  float cand[4][4] = {
    {1.f + tr, m21 - m12, m02 - m20, m10 - m01},
    {m21 - m12, 1.f + m00 - m11 - m22, m01 + m10, m02 + m20},
    {m02 - m20, m01 + m10, 1.f - m00 + m11 - m22, m12 + m21},
    {m10 - m01, m02 + m20, m12 + m21, 1.f - m00 - m11 + m22}};
  int bi = 0; float bd = cand[0][0];
#pragma unroll
  for (int i = 1; i < 4; ++i) { float di = cand[i][i]; if (di > bd) { bd = di; bi = i; } }
#pragma unroll
  for (int j = 0; j < 4; ++j) q[j] = cand[bi][j];
  if (q[0] < 0.f) {
#pragma unroll
    for (int j = 0; j < 4; ++j) q[j] = -q[j];
  }
  normalize4(q);
}
__device__ void tangent_apply(const float* d, const float* t, const float* q, float* tn, float* qn) {
  tn[0] = t[0] + d[0]; tn[1] = t[1] + d[1]; tn[2] = t[2] + d[2];
  float dq[4]; rotvec_to_quat(d + 3, dq);
  quat_mul4(dq, q, qn);
  normalize4(qn);
}

// ---------------------------------------------------------------- point residuals (primal)
__device__ void point_r(const SM* sm, const float* Kc, int p, const float* R, const float* t,
                        float* rx, float* ry) {
  float X0 = sm->x3[p][0], X1 = sm->x3[p][1], X2 = sm->x3[p][2];
  float c0 = R[0] * X0 + R[1] * X1 + R[2] * X2 + t[0];
  float c1 = R[3] * X0 + R[4] * X1 + R[5] * X2 + t[1];
  float c2 = R[6] * X0 + R[7] * X1 + R[8] * X2 + t[2];
  float uu = Kc[0] * c0 + Kc[1] * c1 + Kc[2] * c2;
  float vv = Kc[3] * c0 + Kc[4] * c1 + Kc[5] * c2;
  float ww = Kc[6] * c0 + Kc[7] * c1 + Kc[8] * c2;
  float z = fmaxf(ww, 1e-4f);
  *rx = (uu / z - sm->x2[p][0]) * sm->w2[p][0];
  *ry = (vv / z - sm->x2[p][1]) * sm->w2[p][1];
}
__device__ float point_cost(const SM* sm, const float* Kc, int p, const float* R, const float* t,
                            float delta) {
  float rx, ry; point_r(sm, Kc, p, R, t, &rx, &ry);
  float rn = sqrtf(rx * rx + ry * ry);
  return (rn <= delta) ? 0.5f * rn * rn : delta * (rn - 0.5f * delta);
}
__device__ float point_rw2(const SM* sm, const float* Kc, int p, const float* R, const float* t,
                           float delta) {
  float rx, ry; point_r(sm, Kc, p, R, t, &rx, &ry);
  float rn = sqrtf(rx * rx + ry * ry);
  float wsq = fminf(1.f, delta / fmaxf(rn, 1e-8f));
  return (rx * rx + ry * ry) * wsq;
}
__device__ float robust_cost_block(SM* sm, const float* Kc, const float* t, const float* q,
                                   float delta) {
  float R[9]; quat_to_rot(q, R);
  float c = 0.f;
  for (int p = threadIdx.x; p < NPTS; p += TPB) c += point_cost(sm, Kc, p, R, t, delta);
  return block_sum(sm, c);
}

// ---------------------------------------------------------------- jet residuals (J at d=0)
__device__ void pose_jets(const float* t, const float* q, Jet* tn, Jet* Rj) {
#pragma unroll
  for (int i = 0; i < 3; ++i) { tn[i] = jc(t[i]); tn[i].g[i] = 1.f; }
  // smooth limit of rotvec_to_quat at 0: dq = (1, 0.5*d3, 0.5*d4, 0.5*d5)
  Jet dq0 = jc(1.f), dq1 = jc(0.f), dq2 = jc(0.f), dq3 = jc(0.f);
  dq1.g[3] = 0.5f; dq2.g[4] = 0.5f; dq3.g[5] = 0.5f;
  Jet qm0 = dq0 * q[0] - dq1 * q[1] - dq2 * q[2] - dq3 * q[3];
  Jet qm1 = dq0 * q[1] + dq1 * q[0] + dq2 * q[3] - dq3 * q[2];
  Jet qm2 = dq0 * q[2] - dq1 * q[3] + dq2 * q[0] + dq3 * q[1];
  Jet qm3 = dq0 * q[3] + dq1 * q[2] - dq2 * q[1] + dq3 * q[0];
  Jet n = jsqrt(qm0 * qm0 + qm1 * qm1 + qm2 * qm2 + qm3 * qm3);
  Jet w = jdiv(qm0, n), x = jdiv(qm1, n), y = jdiv(qm2, n), z = jdiv(qm3, n);
  Rj[0] = jc(1.f) - (y * y + z * z) * 2.f; Rj[1] = (x * y - w * z) * 2.f; Rj[2] = (x * z + w * y) * 2.f;
  Rj[3] = (x * y + w * z) * 2.f; Rj[4] = jc(1.f) - (x * x + z * z) * 2.f; Rj[5] = (y * z - w * x) * 2.f;
  Rj[6] = (x * z - w * y) * 2.f; Rj[7] = (y * z + w * x) * 2.f; Rj[8] = jc(1.f) - (x * x + y * y) * 2.f;
}
__device__ void point_rw_jet(const SM* sm, const float* Kc, int p, const Jet* tn, const Jet* Rj,
                             float delta, Jet* ox, Jet* oy) {
  float X0 = sm->x3[p][0], X1 = sm->x3[p][1], X2 = sm->x3[p][2];
  Jet c0 = Rj[0] * X0 + Rj[1] * X1 + Rj[2] * X2 + tn[0];
  Jet c1 = Rj[3] * X0 + Rj[4] * X1 + Rj[5] * X2 + tn[1];
  Jet c2 = Rj[6] * X0 + Rj[7] * X1 + Rj[8] * X2 + tn[2];
  Jet uu = c0 * Kc[0] + c1 * Kc[1] + c2 * Kc[2];
  Jet vv = c0 * Kc[3] + c1 * Kc[4] + c2 * Kc[5];
  Jet ww = c0 * Kc[6] + c1 * Kc[7] + c2 * Kc[8];
  Jet z = jmaxc(ww, 1e-4f);
  Jet rx = (jdiv(uu, z) - sm->x2[p][0]) * sm->w2[p][0];
  Jet ry = (jdiv(vv, z) - sm->x2[p][1]) * sm->w2[p][1];
  Jet rn = jsqrt(rx * rx + ry * ry);
  Jet ratio = jdiv(jc(delta), jmaxc(rn, 1e-8f));
  Jet w = jsqrt(jminc(ratio, 1.f));
  *ox = rx * w; *oy = ry * w;
}

// ---------------------------------------------------------------- 6x6 solve (Gauss, no pivot; H is damped SPD)
__device__ void solve6(float A[6][6], float* bv, float* x) {
  for (int k = 0; k < 6; ++k) {
    float inv = 1.f / A[k][k];
    for (int j = k; j < 6; ++j) A[k][j] *= inv;
    bv[k] *= inv;
    for (int i = k + 1; i < 6; ++i) {
      float f = A[i][k];
      for (int j = k; j < 6; ++j) A[i][j] -= f * A[k][j];
      bv[i] -= f * bv[k];
    }
  }
  for (int i = 5; i >= 0; --i) {
    float s = bv[i];
    for (int j = i + 1; j < 6; ++j) s -= A[i][j] * x[j];
    x[i] = s;
  }
}

// ---------------------------------------------------------------- stateless RNG
__device__ unsigned hash_u(unsigned a, unsigned b, unsigned c) {
  unsigned x = a * 0x9E3779B1u ^ b * 0x85EBCA77u ^ c * 0xC2B2AE3Du;
  x ^= x >> 16; x *= 0x7FEB352Du; x ^= x >> 15; x *= 0x846CA68Bu; x ^= x >> 16;
  return x;
}
__device__ float u01f(unsigned x) { return ((float)(x >> 8) + 0.5f) * (1.f / 16777216.f); }
__device__ float nrmf(unsigned a, unsigned b, unsigned c) {
  float u1 = u01f(hash_u(a, b, c));
  float u2 = u01f(hash_u(a, b, c ^ 0x68BC21EBu));
  return sqrtf(-2.f * logf(fmaxf(u1, 1e-12f))) * cosf(6.28318530718f * u2);
}

// ---------------------------------------------------------------- serial LM (RANSAC, 16-pt subsets)
__device__ void lm_serial(const SM* sm, const float* Kc, const int* idx, float delta, int niter,
                          float* t, float* q) {
  for (int it = 0; it < niter; ++it) {
    Jet tn[3], Rj[9];
    pose_jets(t, q, tn, Rj);
    float H[6][6]; float g[6]; float rsq = 0.f;
#pragma unroll
    for (int i = 0; i < 6; ++i) { g[i] = 0.f; for (int j = 0; j < 6; ++j) H[i][j] = 0.f; }
    for (int i = 0; i < RS_PTS; ++i) {
      Jet ox, oy;
      point_rw_jet(sm, Kc, idx[i], tn, Rj, delta, &ox, &oy);
#pragma unroll
      for (int a = 0; a < 6; ++a) {
        g[a] += ox.g[a] * ox.v + oy.g[a] * oy.v;
#pragma unroll
        for (int b2 = 0; b2 < 6; ++b2) H[a][b2] += ox.g[a] * ox.g[b2] + oy.g[a] * oy.g[b2];
      }
      rsq += ox.v * ox.v + oy.v * oy.v;
    }
    float A[6][6], bv[6], dvec[6];
#pragma unroll
    for (int i = 0; i < 6; ++i) {
      for (int j = 0; j < 6; ++j) A[i][j] = H[i][j];
      A[i][i] += 1e-3f * H[i][i] + 1e-6f;
      bv[i] = -g[i];
    }
    solve6(A, bv, dvec);
    float tn2[3], qn2[4];
    tangent_apply(dvec, t, q, tn2, qn2);
    float Rn[9]; quat_to_rot(qn2, Rn);
    float cn = 0.f;
    for (int i = 0; i < RS_PTS; ++i) cn += point_rw2(sm, Kc, idx[i], Rn, tn2, delta);
    cn *= 0.5f;
    if (cn < 0.5f * rsq) {
#pragma unroll
      for (int j = 0; j < 3; ++j) t[j] = tn2[j];
#pragma unroll
      for (int j = 0; j < 4; ++j) q[j] = qn2[j];
    }
  }
}

// ---------------------------------------------------------------- H = [J r]^T [J r] via WMMA f32 16x16x4
// A(16x4) = (J chunk)^T, B(4x16) = J chunk. Per the CDNA5 VGPR layouts both operands are the
// identical float2 per lane: v[j] = J[rowbase + (lane>>4)*2 + j][lane&15]. Column-major staging
// makes that one aligned ds_load_b64 per K-step. Lanes 8..15 alias columns 0..7 (no masking):
// the duplicated data only lands in D rows/cols 8..15, which are never read. The extracted 7x7
// corner of D holds H(6x6), g = J^T r (col 6), and sum(r^2) at (6,6).
__device__ void build_H_wmma(SM* sm, const float* Kc, const float* t, const float* q, float delta) {
  int tid = threadIdx.x;
  int wave = tid >> 5, lane = tid & 31;
  Jet tn[3], Rj[9];
  pose_jets(t, q, tn, Rj);
  for (int i = lane; i < PPW; i += 32) {
    int p = wave * PPW + i;
    Jet ox, oy;
    point_rw_jet(sm, Kc, p, tn, Rj, delta, &ox, &oy);
#pragma unroll
    for (int a = 0; a < 6; ++a) {
      sm->Jt[wave][a][2 * i]     = ox.g[a];
      sm->Jt[wave][a][2 * i + 1] = oy.g[a];
    }
    sm->Jt[wave][6][2 * i] = ox.v;  sm->Jt[wave][6][2 * i + 1] = oy.v;
    sm->Jt[wave][7][2 * i] = 0.f;   sm->Jt[wave][7][2 * i + 1] = 0.f;
  }
  __syncthreads();

  int m = lane & 15, hf = lane >> 4;
  const v2f* col = (const v2f*)(&sm->Jt[wave][m & 7][0]);  // 8-byte aligned (400 B per column)
  v8f acc = {0.f, 0.f, 0.f, 0.f, 0.f, 0.f, 0.f, 0.f};
#pragma unroll
  for (int c = 0; c < ROWS_PW / 4; ++c) {
    v2f a = col[c * 2 + hf];  // J rows (c*4 + hf*2, c*4 + hf*2 + 1) at this lane's column
    acc = __builtin_amdgcn_wmma_f32_16x16x4_f32(false, a, false, a, (short)0, acc, false, false);
  }
  // D layout: VGPR i, lanes 0-15 -> row M=i, col N=lane
  if (hf == 0 && m < 7) {
#pragma unroll
    for (int mm = 0; mm < 7; ++mm) sm->Hpart[wave][mm][m] = acc[mm];
  }
  __syncthreads();
  if (tid < 49) {
    int mm = tid / 7, nn = tid % 7;
    float s = 0.f;
#pragma unroll
    for (int w = 0; w < NWAVES; ++w) s += sm->Hpart[w][mm][nn];
    sm->Hsum[mm][nn] = s;
  }
  __syncthreads();
}

// ---------------------------------------------------------------- per-sample MC cost (serial over points)
__device__ float sample_cost_one(const SM* sm, const float* Kc, const float* t, const float* q,
                                 const float* d, float delta) {
  float ts[3] = {t[0] + d[0], t[1] + d[1], t[2] + d[2]};
  float dq[4]; rotvec_to_quat(d + 3, dq);
  float qs[4]; quat_mul4(dq, q, qs); normalize4(qs);
  float R[9]; quat_to_rot(qs, R);
  float c = 0.f;
  for (int p = 0; p < NPTS; ++p) c += point_cost(sm, Kc, p, R, ts, delta);
  return c;
}

// ================================================================ main kernel: one block per batch item
__global__ __launch_bounds__(TPB) void epropnp_kernel(const float* __restrict__ gx3,
                                                      const float* __restrict__ gx2,
                                                      const float* __restrict__ gw2,
                                                      const float* __restrict__ gK,
                                                      const float* __restrict__ gE,
                                                      float* __restrict__ per_item_loss) {
  __shared__ SM sm;
  int b = blockIdx.x, tid = threadIdx.x;

  // stage points into LDS
  const float* px3 = gx3 + (size_t)b * NPTS * 3;
  const float* px2 = gx2 + (size_t)b * NPTS * 2;
  const float* pw2 = gw2 + (size_t)b * NPTS * 2;
  for (int i = tid; i < NPTS * 3; i += TPB) ((float*)sm.x3)[i] = px3[i];
  for (int i = tid; i < NPTS * 2; i += TPB) ((float*)sm.x2)[i] = px2[i];
  for (int i = tid; i < NPTS * 2; i += TPB) ((float*)sm.w2)[i] = pw2[i];

  float Kc[9];
#pragma unroll
  for (int j = 0; j < 9; ++j) Kc[j] = gK[b * 9 + j];
  float Rgt[9], tgt[3];
#pragma unroll
  for (int r = 0; r < 3; ++r) {
#pragma unroll
    for (int c = 0; c < 3; ++c) Rgt[r * 3 + c] = gE[b * 12 + r * 4 + c];
    tgt[r] = gE[b * 12 + r * 4 + 3];
  }
  float qgt[4]; rot_to_quat(Rgt, qgt);
  __syncthreads();

  // delta = sqrt(var_x + var_y) * REL_DELTA
  float sx = 0.f, sy = 0.f, sxx = 0.f, syy = 0.f;
  for (int p = tid; p < NPTS; p += TPB) {
    float u = sm.x2[p][0], v = sm.x2[p][1];
    sx += u; sy += v; sxx += u * u; syy += v * v;
  }
  sx = block_sum(&sm, sx); sy = block_sum(&sm, sy);
  sxx = block_sum(&sm, sxx); syy = block_sum(&sm, syy);
  float mxv = sx / NPTS, myv = sy / NPTS;
  float delta = sqrtf(fmaxf(sxx / NPTS - mxv * mxv + syy / NPTS - myv * myv, 0.f)) * REL_DELTA;

  float cost_tgt = robust_cost_block(&sm, Kc, tgt, qgt, delta);

  // ---------------- RANSAC init (4 proposals, serial per thread)
  if (tid < RS_PROPS) {
    int p = tid;
    int idx[RS_PTS];
#pragma unroll
    for (int i = 0; i < RS_PTS; ++i) idx[i] = (int)(hash_u((unsigned)b, 1000u + p, (unsigned)i) % NPTS);
    float tp[3] = {0.5f * nrmf((unsigned)b, 2000u + p, 0u),
                   0.5f * nrmf((unsigned)b, 2000u + p, 1u),
                   4.f + 0.5f * nrmf((unsigned)b, 2000u + p, 2u)};
    float qp[4];
#pragma unroll
    for (int j = 0; j < 4; ++j) qp[j] = nrmf((unsigned)b, 3000u + p, (unsigned)j);
    normalize4(qp);
    lm_serial(&sm, Kc, idx, delta, RS_ITERS_N, tp, qp);
#pragma unroll
    for (int j = 0; j < 3; ++j) sm.cand_t[p][j] = tp[j];
#pragma unroll
    for (int j = 0; j < 4; ++j) sm.cand_q[p][j] = qp[j];
  }
  __syncthreads();

  float cc[RS_PROPS];
  for (int p = 0; p < RS_PROPS; ++p)
    cc[p] = robust_cost_block(&sm, Kc, sm.cand_t[p], sm.cand_q[p], delta);
  int bi = 0;
#pragma unroll
  for (int p = 1; p < RS_PROPS; ++p) if (cc[p] < cc[bi]) bi = p;

  float t[3], q[4];
  if (cc[bi] < cost_tgt) {
#pragma unroll
    for (int j = 0; j < 3; ++j) t[j] = sm.cand_t[bi][j];
#pragma unroll
    for (int j = 0; j < 4; ++j) q[j] = sm.cand_q[bi][j];
  } else {
#pragma unroll
    for (int j = 0; j < 3; ++j) t[j] = tgt[j];
#pragma unroll
    for (int j = 0; j < 4; ++j) q[j] = qgt[j];
  }
  __syncthreads();

  // ---------------- main LM (5 iters, WMMA Gram accumulation)
  for (int it = 0; it < LM_ITERS_N; ++it) {
    build_H_wmma(&sm, Kc, t, q, delta);
    float A[6][6], bv[6], dvec[6];
    float rsq = sm.Hsum[6][6];
#pragma unroll
    for (int i = 0; i < 6; ++i) {
      for (int j = 0; j < 6; ++j) A[i][j] = sm.Hsum[i][j];
      A[i][i] += 1e-3f * sm.Hsum[i][i] + 1e-6f;
      bv[i] = -sm.Hsum[i][6];
    }
    solve6(A, bv, dvec);
    float tn[3], qn[4];
    tangent_apply(dvec, t, q, tn, qn);
    float Rn[9]; quat_to_rot(qn, Rn);
    float local = 0.f;
    for (int p = tid; p < NPTS; p += TPB) local += point_rw2(&sm, Kc, p, Rn, tn, delta);
    float cnew = 0.5f * block_sum(&sm, local);
    if (cnew < 0.5f * rsq) {
#pragma unroll
      for (int j = 0; j < 3; ++j) t[j] = tn[j];
#pragma unroll
      for (int j = 0; j < 4; ++j) q[j] = qn[j];
    }
  }

  // ---------------- hdiag -> sigma init (reuse WMMA Gram diag)
  build_H_wmma(&sm, Kc, t, q, delta);
  float mu[6], sig[6];
#pragma unroll
  for (int j = 0; j < 6; ++j) {
    float hd = sm.Hsum[j][j];
    sig[j] = fminf(fmaxf(rsqrtf(hd + 1e-6f), 1e-3f), 1.f);
    mu[j] = 0.f;
  }
  __syncthreads();

  // ---------------- adaptive multiple importance sampling
  float loss_pred = 0.f;
  for (int k = 0; k < MC_ITERS_N; ++k) {
    if (tid == 0) {
#pragma unroll
      for (int j = 0; j < 6; ++j) { sm.mus[k][j] = mu[j]; sm.sigs[k][j] = sig[j]; }
    }
    // draw MC_PER new samples and cache their cost (costs of old samples are unchanged)
    for (int s = tid; s < MC_PER; s += TPB) {
      int sg = k * MC_PER + s;
      float d[6];
#pragma unroll
      for (int j = 0; j < 6; ++j) {
        d[j] = mu[j] + sig[j] * nrmf((unsigned)b, 4000u + (unsigned)k, (unsigned)(s * 8 + j));
        sm.dall[sg][j] = d[j];
      }
      sm.scost[sg] = sample_cost_one(&sm, Kc, t, q, d, delta);
    }
    __syncthreads();

    int nsamp = (k + 1) * MC_PER;
    float lognm = logf((float)(k + 1));
    for (int s = tid; s < nsamp; s += TPB) {
      float lg[MC_ITERS_N];
      float mxl = -INFINITY;
      for (int j = 0; j <= k; ++j) {
        float acc2 = 0.f, sl = 0.f;
#pragma unroll
        for (int i = 0; i < 6; ++i) {
          float z = (sm.dall[s][i] - sm.mus[j][i]) / sm.sigs[j][i];
          acc2 += z * z;
          sl += logf(sm.sigs[j][i]);
        }
        lg[j] = -0.5f * acc2 - sl - 3.f * LOG2PI_F;
        mxl = fmaxf(mxl, lg[j]);
      }
      float se = 0.f;
      for (int j = 0; j <= k; ++j) se += expf(lg[j] - mxl);
      float logq = mxl + logf(se) - lognm;
      sm.lw[s] = -sm.scost[s] - logq;
    }
    __syncthreads();

    // softmax over samples -> new mu, sigma
    float lmax = -INFINITY;
    for (int s = tid; s < nsamp; s += TPB) lmax = fmaxf(lmax, sm.lw[s]);
    lmax = block_max(&sm, lmax);
    float psum = 0.f;
    for (int s = tid; s < nsamp; s += TPB) psum += expf(sm.lw[s] - lmax);
    float denom = block_sum(&sm, psum);
    if (k == MC_ITERS_N - 1) loss_pred = lmax + logf(denom);  // logsumexp of final logweights
    float newmu[6];
    for (int j = 0; j < 6; ++j) {
      float acc = 0.f;
      for (int s = tid; s < nsamp; s += TPB) acc += expf(sm.lw[s] - lmax) * sm.dall[s][j];
      newmu[j] = block_sum(&sm, acc) / denom;
    }
    for (int j = 0; j < 6; ++j) {
      float acc = 0.f;
      for (int s = tid; s < nsamp; s += TPB) {
        float dd = sm.dall[s][j] - newmu[j];
        acc += expf(sm.lw[s] - lmax) * dd * dd;
      }
      sig[j] = sqrtf(block_sum(&sm, acc) / denom + 1e-8f);
      mu[j] = newmu[j];
    }
    __syncthreads();
  }

  float loss = cost_tgt + loss_pred;
  if (loss != loss) loss = 0.f;  // nan guard (reference: where(isnan, 0, .))
  if (tid == 0) per_item_loss[b] = loss;
}

// ================================================================ finalize: mean(loss)/mean(scale)
__global__ void finalize_kernel(const float* __restrict__ per_item_loss, int nb,
                                const float* __restrict__ scale, int nscale,
                                float* __restrict__ out) {
  if (threadIdx.x == 0 && blockIdx.x == 0) {
    float s = 0.f;
    for (int i = 0; i < nb; ++i) s += per_item_loss[i];
    float sc = 0.f;
    for (int i = 0; i < nscale; ++i) sc += scale[i];
    out[0] = (s / (float)nb) / (sc / (float)nscale);
  }
}

// ================================================================ launch
extern "C" void kernel_launch(void* const* d_in, const int* in_sizes, int n_in,
                              void* d_out, int out_size, void* d_ws, size_t ws_size,
                              hipStream_t stream) {
  const float* x3d = (const float*)d_in[0];
  const float* x2d = (const float*)d_in[1];
  const float* w2d = (const float*)d_in[2];
  const float* K   = (const float*)d_in[3];
  const float* E   = (const float*)d_in[4];
  const float* scale = (const float*)d_in[5];
  int nb = in_sizes[0] / (NPTS * 3);          // 128
  int nscale = in_sizes[5];                   // 10
  float* per_item = (float*)d_ws;             // nb floats of scratch

  epropnp_kernel<<<nb, TPB, 0, stream>>>(x3d, x2d, w2d, K, E, per_item);
  finalize_kernel<<<1, 32, 0, stream>>>(per_item, nb, scale, nscale, (float*)d_out);
}